// MDS_37263136260291
// MI455X (gfx1250) — compile-verified
//
#include <hip/hip_runtime.h>

typedef __attribute__((ext_vector_type(2))) float v2f;
typedef __attribute__((ext_vector_type(8))) float v8f;

#define N_FULL   16384
#define S_SAMP   8192
#define DIM      8
#define DIAG_F   5.0f

// Tile kernel geometry: each wave = one 16x16 tile; 8 waves/block -> block
// covers 16 rows x 128 cols.  Grid = (S/128, S/16) = (64, 512).
#define WAVES_PER_BLOCK 8
#define BLK_THREADS     (WAVES_PER_BLOCK * 32)
#define GRID_X          (S_SAMP / (16 * WAVES_PER_BLOCK))   // 64
#define GRID_Y          (S_SAMP / 16)                        // 512
#define N_PARTIALS      (GRID_X * GRID_Y)                    // 32768

// ---------------------------------------------------------------------------
// Kernel 1: gather z rows by sample_idx, compute squared norms.
// ---------------------------------------------------------------------------
__global__ __launch_bounds__(256)
void mds_gather_z(const float* __restrict__ latent,
                  const int*   __restrict__ idx,
                  float* __restrict__ z_s,
                  float* __restrict__ sq) {
    int t = blockIdx.x * blockDim.x + threadIdx.x;
    if (t >= S_SAMP) return;
    int r = idx[t];
    const float4* src = (const float4*)(latent + (size_t)r * DIM);
    float4 a = src[0];
    float4 b = src[1];
    float4* dst = (float4*)(z_s + (size_t)t * DIM);
    dst[0] = a;
    dst[1] = b;
    sq[t] = a.x*a.x + a.y*a.y + a.z*a.z + a.w*a.w
          + b.x*b.x + b.y*b.y + b.z*b.z + b.w*b.w;
}

// ---------------------------------------------------------------------------
// Kernel 2: per-wave 16x16 Gram tile via V_WMMA_F32_16X16X4_F32 (x2, K=8),
// fused with the (dist - rel)^2 / den accumulation and a deterministic
// per-block reduction into partials[].
//
// A-matrix 16x4 f32 layout (wave32): lane L holds row m = L%16,
//   VGPR0 = K = (L/16)*2, VGPR1 = K = (L/16)*2 + 1.
// B-matrix 4x16 f32 layout (column-major load): lane L holds column n = L%16,
//   same K striping.  Since B = A^T of the column tile, both operands are
//   built identically from z_s rows.
// C/D 16x16 f32 layout: VGPR v, lane L -> D[m = v + 8*(L/16)][n = L%16].
// ---------------------------------------------------------------------------
__global__ __launch_bounds__(BLK_THREADS)
void mds_tile_wmma(const float* __restrict__ relation,
                   const int*   __restrict__ idx,
                   const float* __restrict__ z_s,
                   const float* __restrict__ sq,
                   float* __restrict__ partials) {
    const int lane = threadIdx.x & 31;
    const int wave = threadIdx.x >> 5;
    const int i0 = blockIdx.y * 16;
    const int j0 = (blockIdx.x * WAVES_PER_BLOCK + wave) * 16;

    const int r  = lane & 15;   // m for A, n for B, n for C/D
    const int h  = lane >> 4;   // lane-half selects K group / row-half of D
    const int kb = h * 2;

    const float* zi = z_s + (size_t)(i0 + r) * DIM;
    const float* zj = z_s + (size_t)(j0 + r) * DIM;

    v2f a0 = { zi[kb],     zi[kb + 1] };
    v2f b0 = { zj[kb],     zj[kb + 1] };
    v2f a1 = { zi[kb + 4], zi[kb + 5] };
    v2f b1 = { zj[kb + 4], zj[kb + 5] };

    // Issue the scattered relation loads for this lane's 8 elements *before*
    // the WMMAs so VMEM latency overlaps the matrix pipe.
    const int   j   = j0 + r;          // this lane's column in sampled space
    const int   cj  = idx[j];
    const float sqj = sq[j];

    float relv[8];
    float sqi[8];
    int   ii[8];
#pragma unroll
    for (int v = 0; v < 8; ++v) {
        int i   = i0 + v + 8 * h;
        ii[v]   = i;
        sqi[v]  = sq[i];
        relv[v] = relation[(size_t)idx[i] * N_FULL + (size_t)cj];
    }

    v8f c = {0.f, 0.f, 0.f, 0.f, 0.f, 0.f, 0.f, 0.f};
    c = __builtin_amdgcn_wmma_f32_16x16x4_f32(false, a0, false, b0,
                                              (short)0, c, false, false);
    c = __builtin_amdgcn_wmma_f32_16x16x4_f32(false, a1, false, b1,
                                              (short)0, c, false, false);

    float acc = 0.f;
#pragma unroll
    for (int v = 0; v < 8; ++v) {
        float dot  = c[v];
        float d2   = fmaxf(sqi[v] + sqj - 2.0f * dot, 0.0f);
        float dist = __builtin_sqrtf(d2);
        float rel  = relv[v];
        float den  = (ii[v] == j) ? DIAG_F : rel;
        float diff = dist - rel;
        acc += diff * diff / den;
    }

    // Deterministic block reduction (no float atomics).
    __shared__ float red[BLK_THREADS];
    red[threadIdx.x] = acc;
    __syncthreads();
#pragma unroll
    for (int s = BLK_THREADS / 2; s > 0; s >>= 1) {
        if (threadIdx.x < s) red[threadIdx.x] += red[threadIdx.x + s];
        __syncthreads();
    }
    if (threadIdx.x == 0)
        partials[blockIdx.y * gridDim.x + blockIdx.x] = red[0];
}

// ---------------------------------------------------------------------------
// Kernel 3: deterministic final reduction + sqrt.
// ---------------------------------------------------------------------------
__global__ __launch_bounds__(256)
void mds_reduce(const float* __restrict__ partials, int n,
                float* __restrict__ out) {
    __shared__ float red[256];
    float acc = 0.f;
    for (int i = threadIdx.x; i < n; i += 256)
        acc += partials[i];
    red[threadIdx.x] = acc;
    __syncthreads();
    for (int s = 128; s > 0; s >>= 1) {
        if (threadIdx.x < s) red[threadIdx.x] += red[threadIdx.x + s];
        __syncthreads();
    }
    if (threadIdx.x == 0)
        out[0] = __builtin_sqrtf(red[0]);
}

// ---------------------------------------------------------------------------
extern "C" void kernel_launch(void* const* d_in, const int* in_sizes, int n_in,
                              void* d_out, int out_size, void* d_ws, size_t ws_size,
                              hipStream_t stream) {
    const float* latent   = (const float*)d_in[0];   // [16384, 8]
    const float* relation = (const float*)d_in[1];   // [16384, 16384]
    const int*   sidx     = (const int*)d_in[2];     // [8192]
    float*       out      = (float*)d_out;           // [1]

    // Workspace layout (floats): z_s[S*8] | sq[S] | partials[N_PARTIALS]
    float* wsf      = (float*)d_ws;
    float* z_s      = wsf;
    float* sq       = wsf + (size_t)S_SAMP * DIM;
    float* partials = sq + S_SAMP;

    mds_gather_z<<<(S_SAMP + 255) / 256, 256, 0, stream>>>(latent, sidx, z_s, sq);

    dim3 grid(GRID_X, GRID_Y);
    mds_tile_wmma<<<grid, BLK_THREADS, 0, stream>>>(relation, sidx, z_s, sq,
                                                    partials);

    mds_reduce<<<1, 256, 0, stream>>>(partials, N_PARTIALS, out);
}